// HierarchicalConcepts_76536317215139
// MI455X (gfx1250) — compile-verified
//
#include <hip/hip_runtime.h>
#include <hip/hip_bf16.h>

// ---------------------------------------------------------------------------
// HierarchicalConcepts forward for MI455X (gfx1250, wave32, WMMA).
// GEMMs: V_WMMA_F32_16X16X4_F32, LDS double-buffered via async global->LDS
// (GLOBAL_LOAD_ASYNC_TO_LDS_B128, ASYNCcnt), 128x64 block tile, 8 waves,
// 32x32 per wave (2x2 WMMA subtiles) -> 1.0 LDS load per WMMA.
// ---------------------------------------------------------------------------

#define NL   8192   // NUM_LOW
#define NH   1024   // NUM_HIGH
#define DIMC 2048   // DIM

typedef float v2f __attribute__((ext_vector_type(2)));
typedef float v8f __attribute__((ext_vector_type(8)));

#define LDS_PAD 4                      // row stride 36 dwords = 144B (16B aligned)

// ---- CDNA5 async global->LDS copies (tracked with ASYNCcnt) ----------------
__device__ __forceinline__ unsigned lds_addr32(void* p) {
    return (unsigned)(unsigned long long)(__attribute__((address_space(3))) char*)p;
}
__device__ __forceinline__ void async_copy_b128(void* lds, const void* g) {
    const unsigned l = lds_addr32(lds);
    asm volatile("global_load_async_to_lds_b128 %0, %1, off" :: "v"(l), "v"(g) : "memory");
}
__device__ __forceinline__ void async_copy_b32(void* lds, const void* g) {
    const unsigned l = lds_addr32(lds);
    asm volatile("global_load_async_to_lds_b32 %0, %1, off" :: "v"(l), "v"(g) : "memory");
}

// ---------------------------------------------------------------------------
// WMMA GEMM.
//  NT == true : C[M,N] = alpha * A[M,K] * B[N,K]^T            (B row-major NxK)
//  NT == false: C[M,N] = rs[i] * (A[M,K] * B[K,N]) + addC     (B row-major KxN)
// Block = 256 threads (8 waves), C tile 128x64, K staged 32 at a time with
// double-buffered async loads.
// A-frag (16x4 f32): lane 0-15 -> row M=lane, half selects K {0,1}/{2,3}.
// B-frag (4x16 f32): lane 0-15 -> col N=lane, half selects K {0,1}/{2,3}.
// C/D (16x16 f32):   vgpr v -> M = v + 8*half, N = lane&15.
// ---------------------------------------------------------------------------
template <bool NT>
__global__ __launch_bounds__(256)
void gemm_wmma_f32(const float* __restrict__ A, const float* __restrict__ B,
                   float* __restrict__ C, const float* __restrict__ addC,
                   const float* __restrict__ rowScale,
                   int M, int N, int K, float alpha)
{
    __shared__ float As[2][128][32 + LDS_PAD];
    __shared__ float Bs[2][64][32 + LDS_PAD];

    const int tileM = blockIdx.y * 128;
    const int tileN = blockIdx.x * 64;
    const int t     = threadIdx.x;
    const int wave  = t >> 5;
    const int lane  = t & 31;
    const int half  = lane >> 4;       // K-half selector
    const int lrow  = lane & 15;       // M (A-frag) / N (B-frag) index

    const int sr0 = (wave & 3) * 2;    // M subtiles sr0, sr0+1   (0..7)
    const int sc0 = (wave >> 2) * 2;   // N subtiles sc0, sc0+1   (0..3)

    v8f acc00 = {}, acc01 = {}, acc10 = {}, acc11 = {};

    // ---- async tile staging (no VGPR round-trip, no per-load waits) --------
    auto stage = [&](int buf, int k0) {
        {   // A tile 128x32: 2 threads/row, 4x B128 each
            const int m  = t >> 1;
            const int kb = (t & 1) * 16;
            const float* src = A + (size_t)(tileM + m) * K + k0 + kb;
            float* dst = &As[buf][m][kb];
#pragma unroll
            for (int u = 0; u < 4; ++u)
                async_copy_b128(dst + u * 4, src + u * 4);
        }
        if (NT) {   // B tile 64x32 direct: 4 threads/row, 2x B128 each
            const int n  = t >> 2;
            const int kb = (t & 3) * 8;
            const float* src = B + (size_t)(tileN + n) * K + k0 + kb;
            float* dst = &Bs[buf][n][kb];
#pragma unroll
            for (int u = 0; u < 2; ++u)
                async_copy_b128(dst + u * 4, src + u * 4);
        } else {    // B tile 32(K)x64(N), transposed into Bs[n][kk]: 8x B32
            const int n  = t & 63;
            const int kb = (t >> 6) * 8;
#pragma unroll
            for (int u = 0; u < 8; ++u)
                async_copy_b32(&Bs[buf][n][kb + u],
                               B + (size_t)(k0 + kb + u) * N + tileN + n);
        }
    };

    stage(0, 0);

    int buf = 0;
    for (int k0 = 0; k0 < K; k0 += 32) {
        const bool hasNext = (k0 + 32) < K;
        if (hasNext) stage(buf ^ 1, k0 + 32);   // prefetch next tile

        // Async ops complete in order: waiting "<= N outstanding" (N = ops just
        // issued for the next tile) guarantees the current tile is resident.
        if (hasNext) {
            if (NT) asm volatile("s_wait_asynccnt 0x6" ::: "memory");
            else    asm volatile("s_wait_asynccnt 0xc" ::: "memory");
        } else {
            asm volatile("s_wait_asynccnt 0x0" ::: "memory");
        }
        __syncthreads();

        const float (*Ab)[32 + LDS_PAD] = As[buf];
        const float (*Bb)[32 + LDS_PAD] = Bs[buf];

#pragma unroll
        for (int kk = 0; kk < 32; kk += 4) {
            const int kq = kk + 2 * half;
            v2f a0, a1, b0, b1;
            a0.x = Ab[sr0 * 16 + lrow][kq];
            a0.y = Ab[sr0 * 16 + lrow][kq + 1];
            a1.x = Ab[(sr0 + 1) * 16 + lrow][kq];
            a1.y = Ab[(sr0 + 1) * 16 + lrow][kq + 1];
            b0.x = Bb[sc0 * 16 + lrow][kq];
            b0.y = Bb[sc0 * 16 + lrow][kq + 1];
            b1.x = Bb[(sc0 + 1) * 16 + lrow][kq];
            b1.y = Bb[(sc0 + 1) * 16 + lrow][kq + 1];
            acc00 = __builtin_amdgcn_wmma_f32_16x16x4_f32(
                        false, a0, false, b0, (short)0, acc00, false, false);
            acc01 = __builtin_amdgcn_wmma_f32_16x16x4_f32(
                        false, a0, false, b1, (short)0, acc01, false, false);
            acc10 = __builtin_amdgcn_wmma_f32_16x16x4_f32(
                        false, a1, false, b0, (short)0, acc10, false, false);
            acc11 = __builtin_amdgcn_wmma_f32_16x16x4_f32(
                        false, a1, false, b1, (short)0, acc11, false, false);
        }
        __syncthreads();   // all waves done reading buf before it is re-filled
        buf ^= 1;
    }

    // ---- epilogue: 2x2 subtiles per wave ----
#pragma unroll
    for (int v = 0; v < 8; ++v) {
        const int r0 = tileM + sr0 * 16 + v + 8 * half;
        const int r1 = r0 + 16;
        const int c0 = tileN + sc0 * 16 + lrow;
        const float s0 = alpha * (rowScale ? rowScale[r0] : 1.0f);
        const float s1 = alpha * (rowScale ? rowScale[r1] : 1.0f);
        const size_t i00 = (size_t)r0 * N + c0;
        const size_t i01 = i00 + 16;
        const size_t i10 = (size_t)r1 * N + c0;
        const size_t i11 = i10 + 16;
        float o00 = acc00[v] * s0, o01 = acc01[v] * s0;
        float o10 = acc10[v] * s1, o11 = acc11[v] * s1;
        if (addC) {
            o00 += addC[i00]; o01 += addC[i01];
            o10 += addC[i10]; o11 += addC[i11];
        }
        C[i00] = o00; C[i01] = o01;
        C[i10] = o10; C[i11] = o11;
    }
}

// ---------------------------------------------------------------------------
// Row L2-normalize: normed = w / ||w||, concepts = flag ? normed : w
// One block (256 threads) per row.
// ---------------------------------------------------------------------------
__global__ __launch_bounds__(256)
void row_normalize(const float* __restrict__ W, float* __restrict__ normed,
                   float* __restrict__ concepts, const int* __restrict__ flag,
                   int D)
{
    __shared__ float red[256];
    const int row = blockIdx.x;
    const float* src = W + (size_t)row * D;

    float s = 0.0f;
    for (int i = threadIdx.x; i < D; i += 256) { const float v = src[i]; s += v * v; }
    red[threadIdx.x] = s;
    __syncthreads();
    for (int off = 128; off > 0; off >>= 1) {
        if (threadIdx.x < off) red[threadIdx.x] += red[threadIdx.x + off];
        __syncthreads();
    }
    const float inv = 1.0f / sqrtf(red[0]);
    const bool useNorm = (*flag != 0);
    for (int i = threadIdx.x; i < D; i += 256) {
        const float v  = src[i];
        const float nv = v * inv;
        normed[(size_t)row * D + i]   = nv;
        concepts[(size_t)row * D + i] = useNorm ? nv : v;
    }
}

// ---------------------------------------------------------------------------
// Sparsemax along rows of z[rows, 1024], in place. One 1024-thread block/row.
// Bitonic sort (descending) + inclusive scan in LDS.
// ---------------------------------------------------------------------------
__global__ __launch_bounds__(1024)
void sparsemax_rows(float* __restrict__ z)
{
    __shared__ float s[1024];
    __shared__ float c[1024];
    __shared__ int   ksup;

    const int t = threadIdx.x;
    const size_t base = (size_t)blockIdx.x * 1024;
    const float orig = z[base + t];
    s[t] = orig;
    if (t == 0) ksup = 0;

    // bitonic sort, descending
    for (int k = 2; k <= 1024; k <<= 1) {
        for (int j = k >> 1; j > 0; j >>= 1) {
            __syncthreads();
            const int ixj = t ^ j;
            if (ixj > t) {
                const float a = s[t], b = s[ixj];
                const bool desc = ((t & k) == 0);
                if (desc ? (a < b) : (a > b)) { s[t] = b; s[ixj] = a; }
            }
        }
    }
    __syncthreads();

    // inclusive prefix sum of sorted values
    c[t] = s[t];
    for (int off = 1; off < 1024; off <<= 1) {
        __syncthreads();
        const float v = (t >= off) ? c[t - off] : 0.0f;
        __syncthreads();
        c[t] += v;
    }
    __syncthreads();

    // support: (1 + k*z_sorted[k-1]) > cumsum[k-1], k = t+1
    const bool sup = (1.0f + (float)(t + 1) * s[t]) > c[t];
    if (sup) atomicMax(&ksup, t + 1);
    __syncthreads();

    const int ks = ksup;
    const float tau = (c[ks - 1] - 1.0f) / (float)ks;
    z[base + t] = fmaxf(orig - tau, 0.0f);
}

// ---------------------------------------------------------------------------
// inv[j] = 1 / (sum_i attn[i, j] + 1e-10)   (attn: R x C)
// ---------------------------------------------------------------------------
__global__ __launch_bounds__(256)
void colsum_inv(const float* __restrict__ attn, float* __restrict__ inv,
                int R, int C)
{
    const int j = blockIdx.x * 256 + threadIdx.x;
    if (j >= C) return;
    float s = 0.0f;
    for (int i = 0; i < R; ++i) s += attn[(size_t)i * C + j];
    inv[j] = 1.0f / (s + 1e-10f);
}

// ---------------------------------------------------------------------------
// out[C, R] = in[R, C]^T, 32x32 LDS tiles, 256 threads/block.
// ---------------------------------------------------------------------------
__global__ __launch_bounds__(256)
void transpose32(const float* __restrict__ in, float* __restrict__ out,
                 int R, int C)
{
    __shared__ float tile[32][33];
    const int bx = blockIdx.x * 32;  // column base of `in`
    const int by = blockIdx.y * 32;  // row base of `in`
    const int tx = threadIdx.x & 31;
    const int ty = threadIdx.x >> 5; // 0..7
#pragma unroll
    for (int i = 0; i < 32; i += 8)
        tile[ty + i][tx] = in[(size_t)(by + ty + i) * C + bx + tx];
    __syncthreads();
#pragma unroll
    for (int i = 0; i < 32; i += 8)
        out[(size_t)(bx + ty + i) * R + by + tx] = tile[tx][ty + i];
}

// ---------------------------------------------------------------------------

static inline void gemm_nt(const float* A, const float* B, float* C,
                           int M, int N, int K, float alpha, hipStream_t s)
{
    dim3 g(N / 64, M / 128);
    gemm_wmma_f32<true><<<g, 256, 0, s>>>(A, B, C, nullptr, nullptr, M, N, K, alpha);
}
static inline void gemm_nn(const float* A, const float* B, float* C,
                           const float* addC, const float* rowScale,
                           int M, int N, int K, hipStream_t s)
{
    dim3 g(N / 64, M / 128);
    gemm_wmma_f32<false><<<g, 256, 0, s>>>(A, B, C, addC, rowScale, M, N, K, 1.0f);
}

extern "C" void kernel_launch(void* const* d_in, const int* in_sizes, int n_in,
                              void* d_out, int out_size, void* d_ws, size_t ws_size,
                              hipStream_t stream)
{
    (void)in_sizes; (void)n_in; (void)out_size; (void)ws_size;

    const float* low_w  = (const float*)d_in[0];   // [NL, D]
    const float* high_w = (const float*)d_in[1];   // [NH, D]
    const float* Wq     = (const float*)d_in[2];   // [D, D]
    const float* Wk     = (const float*)d_in[3];   // [D, D]
    const int*   fLow   = (const int*)d_in[4];
    const int*   fHigh  = (const int*)d_in[5];

    // Output layout (flat f32, return order):
    float* o_lowc    = (float*)d_out;                         // [NL, D]
    float* o_lowcos  = o_lowc   + (size_t)NL * DIMC;          // [NL, NL]
    float* o_outhigh = o_lowcos + (size_t)NL * NL;            // [NL, D]
    float* o_highcos = o_outhigh+ (size_t)NL * DIMC;          // [NH, NH]
    float* o_attn    = o_highcos+ (size_t)NH * NH;            // [NL, NH]

    // Workspace layout (f32): ~92 MB with reuse.
    float* w_lown  = (float*)d_ws;                            // [NL, D]; later attn^T [NH, NL]
    float* w_highn = w_lown  + (size_t)NL * DIMC;             // [NH, D]
    float* w_highc = w_highn + (size_t)NH * DIMC;             // [NH, D]
    float* w_k     = w_highc + (size_t)NH * DIMC;             // [NH, D]; later aggregated_low
    float* w_inv   = w_k     + (size_t)NH * DIMC;             // [NH]

    const float scale = 0.022097086912079612f;                // 1/sqrt(2048)

    // 1) Row normalization (+ flag-selected concepts).
    row_normalize<<<NL, 256, 0, stream>>>(low_w,  w_lown,  o_lowc,  fLow,  DIMC);
    row_normalize<<<NH, 256, 0, stream>>>(high_w, w_highn, w_highc, fHigh, DIMC);

    // 2) q = low_concepts @ Wq^T  -> parked in o_lowcos (overwritten later).
    gemm_nt(o_lowc, Wq, o_lowcos, NL, DIMC, DIMC, 1.0f, stream);
    // 3) k = high_concepts @ Wk^T
    gemm_nt(w_highc, Wk, w_k, NH, DIMC, DIMC, 1.0f, stream);
    // 4) logits = (q @ k^T) * scale
    gemm_nt(o_lowcos, w_k, o_attn, NL, NH, DIMC, scale, stream);
    // 5) attn = sparsemax(logits), in place.
    sparsemax_rows<<<NL, 1024, 0, stream>>>(o_attn);

    // 6) cosine-similarity outputs (q no longer needed -> reclaim o_lowcos).
    gemm_nt(w_lown,  w_lown,  o_lowcos,  NL, NL, DIMC, 1.0f, stream);
    gemm_nt(w_highn, w_highn, o_highcos, NH, NH, DIMC, 1.0f, stream);

    // 7) normalize_rows(attn^T) pieces: column sums + explicit transpose
    //    (attn^T reuses the low_normed slab, which is now dead).
    colsum_inv<<<NH / 256, 256, 0, stream>>>(o_attn, w_inv, NL, NH);
    {
        dim3 g(NH / 32, NL / 32);
        transpose32<<<g, 256, 0, stream>>>(o_attn, w_lown, NL, NH);
    }

    // 8) high_related_low = attn @ high_concepts  -> o_outhigh
    gemm_nn(o_attn, w_highc, o_outhigh, nullptr, nullptr, NL, DIMC, NH, stream);
    // 9) aggregated_low = diag(inv) * (attn^T @ low_concepts)  -> reclaims w_k
    gemm_nn(w_lown, o_lowc, w_k, nullptr, w_inv, NH, DIMC, NL, stream);
    // 10) output_high_concepts = high_related_low + attn @ aggregated_low
    //     (in-place add: each element is read-then-written by one thread)
    gemm_nn(o_attn, w_k, o_outhigh, o_outhigh, nullptr, NL, DIMC, NH, stream);
}